// FUGCF_54296976556804
// MI455X (gfx1250) — compile-verified
//
#include <hip/hip_runtime.h>
#include <hip/hip_bf16.h>
#include <cmath>

#define Ucnt 8192
#define Icnt 4096
#define Kd   256
#define NNZ  409600
#define EKEY_SENT 0x7F7F7F7F

typedef float v2f __attribute__((ext_vector_type(2)));
typedef float v8f __attribute__((ext_vector_type(8)));

// order-preserving float -> uint for atomicMin
__device__ __forceinline__ unsigned encMin(float f) {
  unsigned u = __float_as_uint(f);
  return (u & 0x80000000u) ? ~u : (u | 0x80000000u);
}
__device__ __forceinline__ float decMin(unsigned k) {
  unsigned u = (k & 0x80000000u) ? (k & 0x7fffffffu) : ~k;
  return __uint_as_float(u);
}

// ---------------- GEMM: C[M,N] = A[M,K] @ B[K,N] (row-major), wave -> 16x64 tile
__global__ __launch_bounds__(256) void gemm_rr(const float* __restrict__ A,
    const float* __restrict__ B, float* __restrict__ C, int M, int K, int N) {
  const int wave = ((blockIdx.x * 256) + (int)threadIdx.x) >> 5;
  const int lane = threadIdx.x & 31;
  const int ntl = N >> 6;
  const int mt = wave / ntl;
  const int nt = wave - mt * ntl;
  if (mt >= (M >> 4)) return;
  const int m0 = mt << 4, n0 = nt << 6;
  const int rw = lane & 15, kp = lane >> 4;
  const float* Ap  = A + (size_t)(m0 + rw) * K + 2 * kp;
  const float* Bp0 = B + (size_t)(2 * kp) * N + n0 + rw;
  v8f acc0 = {}, acc1 = {}, acc2 = {}, acc3 = {};
  for (int k0 = 0; k0 < K; k0 += 4) {
    v2f a; a.x = Ap[k0]; a.y = Ap[k0 + 1];
    const float* Bp = Bp0 + (size_t)k0 * N;
    v2f b0, b1, b2, b3;
    b0.x = Bp[0];  b0.y = Bp[N];
    b1.x = Bp[16]; b1.y = Bp[N + 16];
    b2.x = Bp[32]; b2.y = Bp[N + 32];
    b3.x = Bp[48]; b3.y = Bp[N + 48];
    acc0 = __builtin_amdgcn_wmma_f32_16x16x4_f32(false, a, false, b0, (short)0, acc0, false, false);
    acc1 = __builtin_amdgcn_wmma_f32_16x16x4_f32(false, a, false, b1, (short)0, acc1, false, false);
    acc2 = __builtin_amdgcn_wmma_f32_16x16x4_f32(false, a, false, b2, (short)0, acc2, false, false);
    acc3 = __builtin_amdgcn_wmma_f32_16x16x4_f32(false, a, false, b3, (short)0, acc3, false, false);
  }
  float* Cp = C + (size_t)(m0 + 8 * kp) * N + n0 + rw;
#pragma unroll
  for (int v = 0; v < 8; ++v) {
    Cp[(size_t)v * N + 0]  = acc0[v];
    Cp[(size_t)v * N + 16] = acc1[v];
    Cp[(size_t)v * N + 32] = acc2[v];
    Cp[(size_t)v * N + 48] = acc3[v];
  }
}

// ---------------- GEMM: C[M,N] = A[M,K] @ Bt[N,K]^T (both row-major)
__global__ __launch_bounds__(256) void gemm_abt(const float* __restrict__ A,
    const float* __restrict__ Bt, float* __restrict__ C, int M, int N, int K) {
  const int wave = ((blockIdx.x * 256) + (int)threadIdx.x) >> 5;
  const int lane = threadIdx.x & 31;
  const int ntl = N >> 6;
  const int mt = wave / ntl;
  const int nt = wave - mt * ntl;
  if (mt >= (M >> 4)) return;
  const int m0 = mt << 4, n0 = nt << 6;
  const int rw = lane & 15, kp = lane >> 4;
  const float* Ap = A  + (size_t)(m0 + rw) * K + 2 * kp;
  const float* B0 = Bt + (size_t)(n0 + rw) * K + 2 * kp;
  const float* B1 = B0 + (size_t)16 * K;
  const float* B2 = B0 + (size_t)32 * K;
  const float* B3 = B0 + (size_t)48 * K;
  v8f acc0 = {}, acc1 = {}, acc2 = {}, acc3 = {};
  for (int k0 = 0; k0 < K; k0 += 4) {
    v2f a;  a.x  = Ap[k0]; a.y  = Ap[k0 + 1];
    v2f b0; b0.x = B0[k0]; b0.y = B0[k0 + 1];
    v2f b1; b1.x = B1[k0]; b1.y = B1[k0 + 1];
    v2f b2; b2.x = B2[k0]; b2.y = B2[k0 + 1];
    v2f b3; b3.x = B3[k0]; b3.y = B3[k0 + 1];
    acc0 = __builtin_amdgcn_wmma_f32_16x16x4_f32(false, a, false, b0, (short)0, acc0, false, false);
    acc1 = __builtin_amdgcn_wmma_f32_16x16x4_f32(false, a, false, b1, (short)0, acc1, false, false);
    acc2 = __builtin_amdgcn_wmma_f32_16x16x4_f32(false, a, false, b2, (short)0, acc2, false, false);
    acc3 = __builtin_amdgcn_wmma_f32_16x16x4_f32(false, a, false, b3, (short)0, acc3, false, false);
  }
  float* Cp = C + (size_t)(m0 + 8 * kp) * N + n0 + rw;
#pragma unroll
  for (int v = 0; v < 8; ++v) {
    Cp[(size_t)v * N + 0]  = acc0[v];
    Cp[(size_t)v * N + 16] = acc1[v];
    Cp[(size_t)v * N + 32] = acc2[v];
    Cp[(size_t)v * N + 48] = acc3[v];
  }
}

// ---------------- mat_expo column stats: min/sum over rows of G @ usf^T (cols < I)
__global__ __launch_bounds__(256) void expo_stats(const float* __restrict__ G,
    const float* __restrict__ usf, unsigned* __restrict__ cmk,
    float* __restrict__ csum) {
  const int wave = ((blockIdx.x * 256) + (int)threadIdx.x) >> 5;
  const int lane = threadIdx.x & 31;
  const int ntl = Icnt >> 6;
  const int mt = wave / ntl;
  const int nt = wave - mt * ntl;
  const int m0 = mt << 4, n0 = nt << 6;
  const int rw = lane & 15, kp = lane >> 4;
  const float* Ap = G   + (size_t)(m0 + rw) * Kd + 2 * kp;
  const float* B0 = usf + (size_t)(n0 + rw) * Kd + 2 * kp;
  const float* B1 = B0 + (size_t)16 * Kd;
  const float* B2 = B0 + (size_t)32 * Kd;
  const float* B3 = B0 + (size_t)48 * Kd;
  v8f acc[4]; acc[0] = v8f{}; acc[1] = v8f{}; acc[2] = v8f{}; acc[3] = v8f{};
  for (int k0 = 0; k0 < Kd; k0 += 4) {
    v2f a;  a.x  = Ap[k0]; a.y  = Ap[k0 + 1];
    v2f b0; b0.x = B0[k0]; b0.y = B0[k0 + 1];
    v2f b1; b1.x = B1[k0]; b1.y = B1[k0 + 1];
    v2f b2; b2.x = B2[k0]; b2.y = B2[k0 + 1];
    v2f b3; b3.x = B3[k0]; b3.y = B3[k0 + 1];
    acc[0] = __builtin_amdgcn_wmma_f32_16x16x4_f32(false, a, false, b0, (short)0, acc[0], false, false);
    acc[1] = __builtin_amdgcn_wmma_f32_16x16x4_f32(false, a, false, b1, (short)0, acc[1], false, false);
    acc[2] = __builtin_amdgcn_wmma_f32_16x16x4_f32(false, a, false, b2, (short)0, acc[2], false, false);
    acc[3] = __builtin_amdgcn_wmma_f32_16x16x4_f32(false, a, false, b3, (short)0, acc[3], false, false);
  }
#pragma unroll
  for (int j = 0; j < 4; ++j) {
    float mn = acc[j][0], sm = acc[j][0];
#pragma unroll
    for (int v = 1; v < 8; ++v) { mn = fminf(mn, acc[j][v]); sm += acc[j][v]; }
    mn = fminf(mn, __shfl_xor(mn, 16));
    sm = sm + __shfl_xor(sm, 16);
    if (lane < 16) {
      const int c = n0 + 16 * j + lane;
      atomicMin(&cmk[c], encMin(mn));
      atomicAdd(&csum[c], sm);
    }
  }
}

// ---------------- elementwise scale kernels
__global__ __launch_bounds__(256) void scale_isf(const float* __restrict__ isf,
    const float* __restrict__ lamp, float* __restrict__ o, int n) {
  int t = blockIdx.x * 256 + threadIdx.x; if (t >= n) return;
  o[t] = isf[t] / lamp[t & (Kd - 1)];
}
__global__ __launch_bounds__(256) void make_usg(const float* __restrict__ uacc,
    const float* __restrict__ usv, float* __restrict__ usg, float invc, int n) {
  int t = blockIdx.x * 256 + threadIdx.x; if (t >= n) return;
  usg[t] = (uacc[t] * invc + usv[t]) * 0.5f;
}
__global__ __launch_bounds__(256) void make_ign(const float* __restrict__ iacc,
    const float* __restrict__ isv, const float* __restrict__ lam,
    float* __restrict__ ign, float invc, int n) {
  int t = blockIdx.x * 256 + threadIdx.x; if (t >= n) return;
  ign[t] = ((iacc[t] * invc + isv[t]) * 0.5f) / lam[t & (Kd - 1)];
}

// ---------------- sparse GNN scatter: item_acc += adj^T @ usv ; user_acc += adj @ isv
__global__ __launch_bounds__(256) void spmm_gnn(const int* __restrict__ rows,
    const int* __restrict__ cols, const float* __restrict__ vals,
    const float* __restrict__ usv, const float* __restrict__ isv,
    float* __restrict__ uacc, float* __restrict__ iacc) {
  int t = blockIdx.x * 256 + threadIdx.x;
  int e = t >> 5; if (e >= NNZ) return;
  int kc = (t & 31) << 3;
  int r = rows[e], c = cols[e]; float v = vals[e];
  const float* us = usv + (size_t)r * Kd + kc;
  const float* is = isv + (size_t)c * Kd + kc;
  float* ua = uacc + (size_t)r * Kd + kc;
  float* ia = iacc + (size_t)c * Kd + kc;
#pragma unroll
  for (int k = 0; k < 8; ++k) {
    atomicAdd(ia + k, v * us[k]);
    atomicAdd(ua + k, v * is[k]);
  }
}

// ---------------- M[I,Kd] = pos_dense^T @ usg (scatter)
__global__ __launch_bounds__(256) void spmm_M(const int* __restrict__ rows,
    const int* __restrict__ cols, const float* __restrict__ vals,
    const float* __restrict__ usg, float* __restrict__ Mm) {
  int t = blockIdx.x * 256 + threadIdx.x;
  int e = t >> 5; if (e >= NNZ) return;
  int kc = (t & 31) << 3;
  int r = rows[e], c = cols[e]; float v = vals[e];
  const float* us = usg + (size_t)r * Kd + kc;
  float* mp = Mm + (size_t)c * Kd + kc;
#pragma unroll
  for (int k = 0; k < 8; ++k) atomicAdd(mp + k, v * us[k]);
}

// ---------------- per-row min encoded key (c*NNZ + e fits in int32)
__global__ __launch_bounds__(256) void ekey_min(const int* __restrict__ rows,
    const int* __restrict__ cols, int* __restrict__ ekey) {
  int e = blockIdx.x * 256 + threadIdx.x; if (e >= NNZ) return;
  atomicMin(&ekey[rows[e]], cols[e] * NNZ + e);
}

// ---------------- apply 0.2 * mat_adjust (2 entries/row), recomputing mat_expo entries
__global__ __launch_bounds__(256) void adjust(const float* __restrict__ G,
    const float* __restrict__ usf, const unsigned* __restrict__ cmk,
    const float* __restrict__ csum, const int* __restrict__ ekey,
    const int* __restrict__ cols, const float* __restrict__ vals,
    float* __restrict__ out) {
  int wave = ((blockIdx.x * 256) + (int)threadIdx.x) >> 5;
  int lane = threadIdx.x & 31;
  int r = wave; if (r >= Ucnt) return;
  int key = ekey[r];
  if (key == EKEY_SENT) return;
  int e = key % NNZ;
  int c1 = cols[e]; float v1 = vals[e];
  int c2 = (c1 + 1) & (Icnt - 1);
  const float* g  = G   + (size_t)r  * Kd;
  const float* u1 = usf + (size_t)c1 * Kd;
  const float* u2 = usf + (size_t)c2 * Kd;
  float d1 = 0.f, d2 = 0.f;
  for (int k = lane; k < Kd; k += 32) {
    float gv = g[k];
    d1 += gv * u1[k];
    d2 += gv * u2[k];
  }
  for (int s = 16; s > 0; s >>= 1) { d1 += __shfl_xor(d1, s); d2 += __shfl_xor(d2, s); }
  if (lane == 0) {
    float mn1 = decMin(cmk[c1]), mn2 = decMin(cmk[c2]);
    float den1 = csum[c1] - (float)Ucnt * mn1 + (float)Ucnt * 1e-8f;
    float den2 = csum[c2] - (float)Ucnt * mn2 + (float)Ucnt * 1e-8f;
    float me1 = (d1 - mn1 + 1e-8f) / den1;
    float me2 = (d2 - mn2 + 1e-8f) / den2;
    float dd1 = (v1 == 0.f) ? 1e-5f : v1;
    float dd2 = (v1 == 0.f) ? 1e-5f : -v1;
    out[(size_t)r * Icnt + c1] += 0.2f * (me1 / dd1);
    out[(size_t)r * Icnt + c2] += 0.2f * (me2 / dd2);
  }
}

extern "C" void kernel_launch(void* const* d_in, const int* in_sizes, int n_in,
                              void* d_out, int out_size, void* d_ws, size_t ws_size,
                              hipStream_t stream) {
  const float* adj  = (const float*)d_in[0];
  const float* nadj = (const float*)d_in[1];
  const float* usv  = (const float*)d_in[2];
  const float* isv  = (const float*)d_in[3];
  const float* usf  = (const float*)d_in[4];
  const float* isf  = (const float*)d_in[5];
  const int*   prow = (const int*)d_in[6];
  const int*   pcol = (const int*)d_in[7];
  const float* pval = (const float*)d_in[8];
  const float* lam  = (const float*)d_in[9];
  const float* lamp = (const float*)d_in[10];
  float* out = (float*)d_out;

  char* ws = (char*)d_ws;
  size_t off = 0;
  auto alloc = [&](size_t bytes) -> void* {
    void* p = ws + off;
    off += (bytes + 255) & ~(size_t)255;
    return p;
  };
  float*    G    = (float*)alloc((size_t)Ucnt * Kd * 4);
  float*    Hh   = (float*)alloc((size_t)Ucnt * Kd * 4);
  float*    isfS = (float*)alloc((size_t)Icnt * Kd * 4);
  float*    ignS = (float*)alloc((size_t)Icnt * Kd * 4);
  float*    usg  = (float*)alloc((size_t)Ucnt * Kd * 4);
  float*    iacc = (float*)alloc((size_t)Icnt * Kd * 4);
  float*    uacc = (float*)alloc((size_t)Ucnt * Kd * 4);
  float*    Mm   = (float*)alloc((size_t)Icnt * Kd * 4);
  unsigned* cmk  = (unsigned*)alloc((size_t)Icnt * 4);
  float*    csum = (float*)alloc((size_t)Icnt * 4);
  int*      ek   = (int*)alloc((size_t)Ucnt * 4);

  hipMemsetAsync(iacc, 0, (size_t)Icnt * Kd * 4, stream);
  hipMemsetAsync(uacc, 0, (size_t)Ucnt * Kd * 4, stream);
  hipMemsetAsync(Mm,   0, (size_t)Icnt * Kd * 4, stream);
  hipMemsetAsync(csum, 0, (size_t)Icnt * 4, stream);
  hipMemsetAsync(cmk, 0xFF, (size_t)Icnt * 4, stream);
  hipMemsetAsync(ek, 0x7F, (size_t)Ucnt * 4, stream);

  const float invc = 1.0f / sqrtf((float)Ucnt * (float)Icnt);

  scale_isf<<<(Icnt * Kd) / 256, 256, 0, stream>>>(isf, lamp, isfS, Icnt * Kd);

  // G = adj @ isfS : M=U, K=I, N=Kd -> 2048 waves -> 256 blocks
  gemm_rr<<<(Ucnt / 16) * (Kd / 64) / 8, 256, 0, stream>>>(adj, isfS, G, Ucnt, Icnt, Kd);

  // column min/sum of mat_expo columns [0, I): 32768 waves -> 4096 blocks
  expo_stats<<<(Ucnt / 16) * (Icnt / 64) / 8, 256, 0, stream>>>(G, usf, cmk, csum);

  // sparse GNN smoothing scatters
  spmm_gnn<<<(NNZ * 32) / 256, 256, 0, stream>>>(prow, pcol, pval, usv, isv, uacc, iacc);
  make_usg<<<(Ucnt * Kd) / 256, 256, 0, stream>>>(uacc, usv, usg, invc, Ucnt * Kd);
  make_ign<<<(Icnt * Kd) / 256, 256, 0, stream>>>(iacc, isv, lam, ignS, invc, Icnt * Kd);

  // M = pos_dense^T @ usg
  spmm_M<<<(NNZ * 32) / 256, 256, 0, stream>>>(prow, pcol, pval, usg, Mm);

  // H = norm_adj @ ignS
  gemm_rr<<<(Ucnt / 16) * (Kd / 64) / 8, 256, 0, stream>>>(nadj, ignS, Hh, Ucnt, Icnt, Kd);

  // out = H @ Mm^T : M=U, N=I, K=Kd
  gemm_abt<<<(Ucnt / 16) * (Icnt / 64) / 8, 256, 0, stream>>>(Hh, Mm, out, Ucnt, Icnt, Kd);

  // compute_P adjustments
  ekey_min<<<NNZ / 256, 256, 0, stream>>>(prow, pcol, ek);
  adjust<<<Ucnt / 8, 256, 0, stream>>>(G, usf, cmk, csum, ek, pcol, pval, out);
}